// ResNET_WL_84155589198212
// MI455X (gfx1250) — compile-verified
//
#include <hip/hip_runtime.h>

// MI455X / gfx1250, wave32. fp32 WMMA (V_WMMA_F32_16X16X4_F32) for all GEMMs.

typedef __attribute__((ext_vector_type(2))) float v2f;
typedef __attribute__((ext_vector_type(8))) float v8f;

__device__ __forceinline__ v8f wmma_f32(v2f a, v2f b, v8f c) {
  // 8 args: (neg_a, A, neg_b, B, c_mod, C, reuse_a, reuse_b)
  return __builtin_amdgcn_wmma_f32_16x16x4_f32(false, a, false, b, (short)0, c,
                                               false, false);
}

// ---------------------------------------------------------------------------
// Generic fp32 WMMA GEMM: C = act(A*W [+ A2*W2] + bias)
// One 16x16 output tile per wave; 8 waves/block share one 16-row A slab in LDS
// (row stride padded to K+4 -> conflict-free ds_load_b64 across 64 banks).
// K-loop unrolled x4 (16 K per iter): 8 coalesced global b32 loads + 4 LDS b64
// loads feed 4 back-to-back v_wmma. B addressing = one base VGPR + immediate
// offsets (compile-time N), advanced by 16*N per iteration.
// ---------------------------------------------------------------------------
template <int K, int N, bool DUAL, bool RELU>
__launch_bounds__(256)
__global__ void gemm16_kernel(const float* __restrict__ A,
                              const float* __restrict__ W,
                              const float* __restrict__ A2,
                              const float* __restrict__ W2,
                              const float* __restrict__ bias,
                              float* __restrict__ C) {
  constexpr int STRIDE = K + 4;  // LDS row pad: lane l starts at bank (4*l)%64
  __shared__ float lds[(DUAL ? 32 : 16) * STRIDE];
  const int tid  = threadIdx.x;
  const int wave = tid >> 5;
  const int lane = tid & 31;
  const int kh   = (lane >> 4) << 1;  // K sub-offset: 0 (lanes 0-15) / 2 (16-31)
  const int ln   = lane & 15;
  constexpr int bpr = (N >> 4) >> 3;  // blocks per 16-row slab
  const int tm   = blockIdx.x / bpr;
  const int tn   = (blockIdx.x % bpr) * 8 + wave;
  const int row0 = tm << 4;
  const int col  = (tn << 4) + ln;

  for (int i = tid; i < 16 * K; i += 256) {
    const int r = i / K, c = i % K;  // K is a power of two -> shift/mask
    lds[r * STRIDE + c] = A[(size_t)(row0 + r) * K + c];
  }
  if (DUAL) {
    for (int i = tid; i < 16 * K; i += 256) {
      const int r = i / K, c = i % K;
      lds[16 * STRIDE + r * STRIDE + c] = A2[(size_t)(row0 + r) * K + c];
    }
  }
  __syncthreads();

  v8f acc = {};
#pragma unroll
  for (int p = 0; p < (DUAL ? 2 : 1); ++p) {
    const int la = (p ? 16 * STRIDE : 0) + ln * STRIDE + kh;
    const float* __restrict__ wp = (p ? W2 : W) + (size_t)kh * N + col;
    for (int k0 = 0; k0 < K; k0 += 16) {
      __builtin_prefetch(wp + 16 * N, 0, 1);  // speculative: drop-on-fault safe
      v2f b0, b1, b2, b3;
      b0.x = wp[0];      b0.y = wp[N];
      b1.x = wp[4 * N];  b1.y = wp[5 * N];
      b2.x = wp[8 * N];  b2.y = wp[9 * N];
      b3.x = wp[12 * N]; b3.y = wp[13 * N];
      v2f a0 = *(const v2f*)&lds[la + k0];
      v2f a1 = *(const v2f*)&lds[la + k0 + 4];
      v2f a2 = *(const v2f*)&lds[la + k0 + 8];
      v2f a3 = *(const v2f*)&lds[la + k0 + 12];
      acc = wmma_f32(a0, b0, acc);
      acc = wmma_f32(a1, b1, acc);
      acc = wmma_f32(a2, b2, acc);
      acc = wmma_f32(a3, b3, acc);
      wp += 16 * N;
    }
  }

  const int mh = (lane >> 4) << 3;  // C/D layout: m = r + 8*(lane>=16)
  const float bv = bias[col];
#pragma unroll
  for (int r = 0; r < 8; ++r) {
    float v = acc[r] + bv;
    if (RELU) v = fmaxf(v, 0.0f);
    C[(size_t)(row0 + mh + r) * N + col] = v;
  }
}

// ---------------------------------------------------------------------------
// Row squared norms of xie ([8192,256])
// ---------------------------------------------------------------------------
__global__ void norms_kernel(const float* __restrict__ xie, float* __restrict__ x2) {
  const int row = blockIdx.x * blockDim.x + threadIdx.x;  // 8192 rows
  const float* p = xie + ((size_t)row << 8);
  float s = 0.f;
#pragma unroll 8
  for (int i = 0; i < 256; ++i) s += p[i] * p[i];
  x2[row] = s;
}

// ---------------------------------------------------------------------------
// Per-image pairwise squared distances via WMMA Gram matrix:
//   d2[m][n] = |xm|^2 + |xn|^2 - 2*<xm,xn>  (+1e30 on diagonal)
// B = X^T read as contiguous v2f per lane (single base + immediate offsets).
// grid = 32 images * 16 row-tiles * 2 col-halves = 1024 blocks
// ---------------------------------------------------------------------------
__launch_bounds__(256)
__global__ void knn_dist_kernel(const float* __restrict__ xie,
                                const float* __restrict__ x2,
                                float* __restrict__ d2) {
  constexpr int STRIDE = 260;  // 256 + 4 pad
  __shared__ float lds[16 * STRIDE];
  const int tid  = threadIdx.x;
  const int wave = tid >> 5;
  const int lane = tid & 31;
  const int kh   = (lane >> 4) << 1;
  const int ln   = lane & 15;
  const int img  = blockIdx.x >> 5;
  const int rem  = blockIdx.x & 31;
  const int row0 = (rem >> 1) << 4;
  const int tn   = ((rem & 1) << 3) + wave;
  const int coln = (tn << 4) + ln;
  const float* __restrict__ Xb = xie + ((size_t)img << 16);

  for (int i = tid; i < 4096; i += 256)
    lds[(i >> 8) * STRIDE + (i & 255)] = Xb[((row0 + (i >> 8)) << 8) + (i & 255)];
  __syncthreads();

  const int la = ln * STRIDE + kh;
  const float* __restrict__ xp = Xb + ((size_t)coln << 8) + kh;
  v8f acc = {};
  for (int k0 = 0; k0 < 256; k0 += 16) {
    v2f b0 = *(const v2f*)&xp[k0];
    v2f b1 = *(const v2f*)&xp[k0 + 4];
    v2f b2 = *(const v2f*)&xp[k0 + 8];
    v2f b3 = *(const v2f*)&xp[k0 + 12];
    v2f a0 = *(const v2f*)&lds[la + k0];
    v2f a1 = *(const v2f*)&lds[la + k0 + 4];
    v2f a2 = *(const v2f*)&lds[la + k0 + 8];
    v2f a3 = *(const v2f*)&lds[la + k0 + 12];
    acc = wmma_f32(a0, b0, acc);
    acc = wmma_f32(a1, b1, acc);
    acc = wmma_f32(a2, b2, acc);
    acc = wmma_f32(a3, b3, acc);
  }

  const int mh = (lane >> 4) << 3;
  const float xn = x2[(img << 8) + coln];
#pragma unroll
  for (int r = 0; r < 8; ++r) {
    const int m = row0 + mh + r;
    float dd = x2[(img << 8) + m] + xn - 2.0f * acc[r];
    if (m == coln) dd += 1e30f;  // exclude self
    d2[((size_t)img << 16) + ((size_t)m << 8) + coln] = dd;
  }
}

// ---------------------------------------------------------------------------
// Register-resident insertion top-10 smallest per row (ties keep lower index,
// matching jax.lax.top_k).
// ---------------------------------------------------------------------------
__global__ void topk_kernel(const float* __restrict__ d2, int* __restrict__ nbr) {
  const int row = blockIdx.x * blockDim.x + threadIdx.x;  // 8192 rows
  const float* p = d2 + ((size_t)row << 8);
  float bd[10];
  int bi[10];
#pragma unroll
  for (int i = 0; i < 10; ++i) { bd[i] = 3.4e38f; bi[i] = 0; }
  for (int j = 0; j < 256; ++j) {
    const float d = p[j];
    if (d < bd[9]) {
      bd[9] = d; bi[9] = j;
#pragma unroll
      for (int t = 9; t > 0; --t) {
        if (bd[t] < bd[t - 1]) {
          float td = bd[t]; bd[t] = bd[t - 1]; bd[t - 1] = td;
          int   ti = bi[t]; bi[t] = bi[t - 1]; bi[t - 1] = ti;
        }
      }
    }
  }
#pragma unroll
  for (int i = 0; i < 10; ++i) nbr[row * 10 + i] = bi[i];
}

// ---------------------------------------------------------------------------
// Mean neighbor aggregation: one block per node row, coalesced across channels.
// ---------------------------------------------------------------------------
__global__ void agg_kernel(const float* __restrict__ xie,
                           const int* __restrict__ nbr,
                           float* __restrict__ agg) {
  const int row = blockIdx.x;   // 8192
  const int c   = threadIdx.x;  // 256
  const int img = row >> 8;
  const int* nb = nbr + row * 10;
  float s = 0.f;
#pragma unroll
  for (int t = 0; t < 10; ++t)
    s += xie[(((size_t)(img << 8) + nb[t]) << 8) + c];
  agg[((size_t)row << 8) + c] = s * 0.1f;
}

// ---------------------------------------------------------------------------
// Split-K WMMA for h = gep_flat[32,65536] @ W1[65536,512]
// 128 K-chunks of 512; grid = 128 chunks * 8 blocks (2 row-tiles * 4 col-groups)
// Deterministic: partials to hp, reduced in a second pass (no float atomics).
// ---------------------------------------------------------------------------
__launch_bounds__(256)
__global__ void splitk_kernel(const float* __restrict__ A,
                              const float* __restrict__ W1,
                              float* __restrict__ hp) {
  constexpr int STRIDE = 516;  // 512 + 4 pad
  constexpr int N = 512;
  __shared__ float lds[16 * STRIDE];
  const int tid   = threadIdx.x;
  const int wave  = tid >> 5;
  const int lane  = tid & 31;
  const int kh    = (lane >> 4) << 1;
  const int ln    = lane & 15;
  const int chunk = blockIdx.x >> 3;
  const int rem   = blockIdx.x & 7;
  const int row0  = (rem >> 2) << 4;
  const int tn    = ((rem & 3) << 3) + wave;
  const int col   = (tn << 4) + ln;
  const int kBase = chunk << 9;

  for (int i = tid; i < 8192; i += 256)
    lds[(i >> 9) * STRIDE + (i & 511)] =
        A[((size_t)(row0 + (i >> 9)) << 16) + kBase + (i & 511)];
  __syncthreads();

  const int la = ln * STRIDE + kh;
  const float* __restrict__ wp = W1 + (((size_t)kBase + kh) << 9) + col;
  v8f acc = {};
  for (int k0 = 0; k0 < 512; k0 += 16) {
    __builtin_prefetch(wp + 16 * N, 0, 1);
    v2f b0, b1, b2, b3;
    b0.x = wp[0];      b0.y = wp[N];
    b1.x = wp[4 * N];  b1.y = wp[5 * N];
    b2.x = wp[8 * N];  b2.y = wp[9 * N];
    b3.x = wp[12 * N]; b3.y = wp[13 * N];
    v2f a0 = *(const v2f*)&lds[la + k0];
    v2f a1 = *(const v2f*)&lds[la + k0 + 4];
    v2f a2 = *(const v2f*)&lds[la + k0 + 8];
    v2f a3 = *(const v2f*)&lds[la + k0 + 12];
    acc = wmma_f32(a0, b0, acc);
    acc = wmma_f32(a1, b1, acc);
    acc = wmma_f32(a2, b2, acc);
    acc = wmma_f32(a3, b3, acc);
    wp += 16 * N;
  }
  __builtin_amdgcn_s_wait_tensorcnt(0);  // CDNA5 split-counter path (no-op here)

  const int mh = (lane >> 4) << 3;
#pragma unroll
  for (int r = 0; r < 8; ++r)
    hp[((size_t)chunk << 14) + ((size_t)(row0 + mh + r) << 9) + col] = acc[r];
}

// h[b,j] = relu(b1[j] + sum_chunks hp[c][b][j])  -- deterministic reduction
__global__ void reduceh_kernel(const float* __restrict__ hp,
                               const float* __restrict__ b1,
                               float* __restrict__ h) {
  const int i = blockIdx.x * blockDim.x + threadIdx.x;  // 16384
  float s = b1[i & 511];
  for (int c = 0; c < 128; ++c) s += hp[((size_t)c << 14) + i];
  h[i] = fmaxf(s, 0.f);
}

// out[b] = sum_j h[b,j]*W2[j] + b2 ; LDS tree reduction, one block per batch row
__global__ void final_kernel(const float* __restrict__ h,
                             const float* __restrict__ W2,
                             const float* __restrict__ b2,
                             float* __restrict__ out) {
  __shared__ float red[256];
  const int b = blockIdx.x, t = threadIdx.x;
  red[t] = h[(b << 9) + t] * W2[t] + h[(b << 9) + t + 256] * W2[t + 256];
  __syncthreads();
  for (int off = 128; off > 0; off >>= 1) {
    if (t < off) red[t] += red[t + off];
    __syncthreads();
  }
  if (t == 0) out[b] = red[0] + b2[0];
}

// ---------------------------------------------------------------------------
extern "C" void kernel_launch(void* const* d_in, const int* in_sizes, int n_in,
                              void* d_out, int out_size, void* d_ws, size_t ws_size,
                              hipStream_t stream) {
  (void)in_sizes; (void)n_in; (void)out_size; (void)ws_size;
  const float* xi     = (const float*)d_in[0];  // [32,256,512]
  const float* W_img  = (const float*)d_in[1];  // [512,256]
  const float* b_img  = (const float*)d_in[2];  // [256]
  const float* W_self = (const float*)d_in[3];  // [256,256]
  const float* W_nbr  = (const float*)d_in[4];  // [256,256]
  const float* b_gnn  = (const float*)d_in[5];  // [256]
  const float* W1     = (const float*)d_in[6];  // [65536,512]
  const float* b1     = (const float*)d_in[7];  // [512]
  const float* W2     = (const float*)d_in[8];  // [512,1]
  const float* b2     = (const float*)d_in[9];  // [1]
  // d_in[10] = k (always 10)

  float* outp = (float*)d_out;
  float* xie  = outp;            // [8192,256]
  float* gep  = outp + 2097152;  // [8192,256]
  float* yout = outp + 4194304;  // [32]

  float* ws  = (float*)d_ws;
  float* G   = ws;                    // d2     [32,256,256]  8 MB
  float* x2  = ws + 2097152;          // norms  [8192]
  float* agg = ws + 2105344;          // agg    [8192,256]    8 MB
  int*   nbr = (int*)(ws + 4202496);  // idx    [8192,10]
  float* hp  = ws + 4284416;          // splitK [128,32,512]  8 MB
  float* h   = ws + 6381568;          // h      [32,512]

  // 1) xie = xi @ W_img + b_img          (WMMA, M=8192 K=512 N=256)
  gemm16_kernel<512, 256, false, false>
      <<<1024, 256, 0, stream>>>(xi, W_img, nullptr, nullptr, b_img, xie);
  // 2) row norms
  norms_kernel<<<32, 256, 0, stream>>>(xie, x2);
  // 3) pairwise d2 per image             (WMMA Gram)
  knn_dist_kernel<<<1024, 256, 0, stream>>>(xie, x2, G);
  // 4) top-10 neighbors per node
  topk_kernel<<<32, 256, 0, stream>>>(G, nbr);
  // 5) mean aggregation
  agg_kernel<<<8192, 256, 0, stream>>>(xie, nbr, agg);
  // 6) gep = relu(xie@W_self + agg@W_nbr + b_gnn)   (fused dual WMMA GEMM)
  gemm16_kernel<256, 256, true, true>
      <<<1024, 256, 0, stream>>>(xie, W_self, agg, W_nbr, b_gnn, gep);
  // 7) split-K partials for h            (WMMA, K=65536 over 128 chunks)
  splitk_kernel<<<1024, 256, 0, stream>>>(gep, W1, hp);
  // 8) h = relu(sum partials + b1)
  reduceh_kernel<<<64, 256, 0, stream>>>(hp, b1, h);
  // 9) out = h @ W2 + b2
  final_kernel<<<32, 256, 0, stream>>>(h, W2, b2, yout);
}